// ConvexHullModel_28922309771286
// MI455X (gfx1250) — compile-verified
//
#include <hip/hip_runtime.h>
#include <hip/hip_bf16.h>

typedef __attribute__((ext_vector_type(2))) float v2f;
typedef __attribute__((ext_vector_type(8))) float v8f;

#define N_NODES 200000
#define N_EDGES 6400000
#define N_GRAPHS 512

// ---------------------------------------------------------------------------
// helpers
// ---------------------------------------------------------------------------
__device__ __forceinline__ float ldA(const float* __restrict__ p, int m, int k, int in_dim) {
    return (k < in_dim) ? p[m * in_dim + k] : 0.0f;
}
// W is [OUT, IN] row-major; B fragment wants W^T i.e. B[k][n] = W[n][k]
__device__ __forceinline__ float ldB(const float* __restrict__ W, int n, int k, int in_dim, int out_dim) {
    return (k < in_dim && n < out_dim) ? W[n * in_dim + k] : 0.0f;
}

__global__ void zero_f32(float* __restrict__ p, int count) {
    int i = blockIdx.x * blockDim.x + threadIdx.x;
    if (i < count) p[i] = 0.0f;
}

// ---------------------------------------------------------------------------
// Edge scatter-add: agg[dst] += x[src].  Node tables fit in the 192MB L2, so
// the random gathers + global_atomic_add_f32 stay on-chip; HBM traffic is
// dominated by streaming the int64 edge list.
// ---------------------------------------------------------------------------
__global__ void scatter_add3(const float* __restrict__ x,
                             const long long* __restrict__ src,
                             const long long* __restrict__ dst,
                             float* __restrict__ agg, int n_edges) {
    int e = blockIdx.x * blockDim.x + threadIdx.x;
    if (e >= n_edges) return;
    int s = (int)src[e];
    int d = (int)dst[e];
    atomicAdd(&agg[d * 3 + 0], x[s * 3 + 0]);
    atomicAdd(&agg[d * 3 + 1], x[s * 3 + 1]);
    atomicAdd(&agg[d * 3 + 2], x[s * 3 + 2]);
}

template <int IN>  // IN multiple of 4; one thread per (edge, 4-channel chunk)
__global__ void scatter_add_vec(const float* __restrict__ x,
                                const long long* __restrict__ src,
                                const long long* __restrict__ dst,
                                float* __restrict__ agg, int n_edges) {
    constexpr int CH = IN / 4;
    long long gid = (long long)blockIdx.x * blockDim.x + threadIdx.x;
    if (gid >= (long long)n_edges * CH) return;
    int e  = (int)(gid / CH);
    int c4 = (int)(gid % CH) * 4;
    int s = (int)src[e];
    int d = (int)dst[e];
    const float4 v = *(const float4*)(x + s * IN + c4);
    float* a = agg + d * IN + c4;
    atomicAdd(a + 0, v.x);
    atomicAdd(a + 1, v.y);
    atomicAdd(a + 2, v.z);
    atomicAdd(a + 3, v.w);
}

// ---------------------------------------------------------------------------
// Node transform with fp32 WMMA:
//   out = relu(agg @ Wrel^T + b + xin @ Wroot^T)
// One wave per (16-node tile, 16-out-channel tile). K accumulated in blocks
// of 4 via V_WMMA_F32_16X16X4_F32, chaining two WMMAs (rel + root) into one
// f32 accumulator.
//
// A 16x4 f32 fragment (ISA 7.12.2): lane L holds row m = mbase + (L&15);
//   VGPR0 = A[m][kb + 2*half], VGPR1 = A[m][kb + 2*half + 1], half = L>>4.
// B 4x16 f32 fragment (mirror): lane L holds col n = nbase + (L&15);
//   VGPR0 = B[kb + 2*half][n], VGPR1 = B[kb + 2*half + 1][n].
// C/D 16x16 f32: c[r] @ lane L = D[mbase + r + 8*half][nbase + (L&15)].
// ---------------------------------------------------------------------------
template <int IN, int OUT>
__global__ void __launch_bounds__(256) transform_relu_wmma(
    const float* __restrict__ agg, const float* __restrict__ xin,
    const float* __restrict__ Wrel, const float* __restrict__ bias,
    const float* __restrict__ Wroot, float* __restrict__ out, int n_nodes) {
    constexpr int NT = (OUT + 15) / 16;
    int wave = (blockIdx.x * blockDim.x + threadIdx.x) >> 5;
    int lane = threadIdx.x & 31;
    int m_tiles = n_nodes >> 4;            // 200000/16 = 12500 exact
    if (wave >= m_tiles * NT) return;      // wave-uniform: EXEC stays all-1s
    int mt = wave / NT, nt = wave % NT;
    int mbase = mt << 4;
    int half = lane >> 4, lo = lane & 15;
    int mA = mbase + lo;                   // A-fragment row for this lane
    int n  = nt * 16 + lo;                 // B/C column for this lane

    v8f c = {0.f, 0.f, 0.f, 0.f, 0.f, 0.f, 0.f, 0.f};
    for (int kb = 0; kb < IN; kb += 4) {
        int k0 = kb + 2 * half;
        int k1 = k0 + 1;
        v2f a1, b1, a2, b2;
        a1.x = ldA(agg, mA, k0, IN);           a1.y = ldA(agg, mA, k1, IN);
        b1.x = ldB(Wrel, n, k0, IN, OUT);      b1.y = ldB(Wrel, n, k1, IN, OUT);
        a2.x = ldA(xin, mA, k0, IN);           a2.y = ldA(xin, mA, k1, IN);
        b2.x = ldB(Wroot, n, k0, IN, OUT);     b2.y = ldB(Wroot, n, k1, IN, OUT);
        c = __builtin_amdgcn_wmma_f32_16x16x4_f32(false, a1, false, b1, (short)0, c, false, false);
        c = __builtin_amdgcn_wmma_f32_16x16x4_f32(false, a2, false, b2, (short)0, c, false, false);
    }

    if (n < OUT) {
        float bv = bias[n];
        #pragma unroll
        for (int r = 0; r < 8; ++r) {
            int m = mbase + r + half * 8;
            float v = c[r] + bv;
            out[m * OUT + n] = v > 0.0f ? v : 0.0f;
        }
    }
}

// ---------------------------------------------------------------------------
// segment_max pool: h >= 0 after relu, so uint bit-pattern atomicMax == float
// max with 0.0f init.
// ---------------------------------------------------------------------------
__global__ void pool_max8(const float* __restrict__ h,
                          const long long* __restrict__ batch,
                          unsigned int* __restrict__ pooled, int n_nodes) {
    int gid = blockIdx.x * blockDim.x + threadIdx.x;
    if (gid >= n_nodes * 8) return;
    int node = gid >> 3;
    int cc   = gid & 7;
    int g = (int)batch[node];
    atomicMax(&pooled[g * 8 + cc], __float_as_uint(h[gid]));
}

__global__ void final_linear(const float* __restrict__ pooled,
                             const float* __restrict__ Wlin,
                             const float* __restrict__ blin,
                             float* __restrict__ out, int n_graphs) {
    int g = blockIdx.x * blockDim.x + threadIdx.x;
    if (g >= n_graphs) return;
    float s = blin[0];
    #pragma unroll
    for (int c = 0; c < 8; ++c) s += pooled[g * 8 + c] * Wlin[c];
    out[g] = s;
}

// ---------------------------------------------------------------------------
// launch
// ---------------------------------------------------------------------------
static inline size_t alignup(size_t x) { return (x + 255) & ~(size_t)255; }

extern "C" void kernel_launch(void* const* d_in, const int* in_sizes, int n_in,
                              void* d_out, int out_size, void* d_ws, size_t ws_size,
                              hipStream_t stream) {
    const float*     x     = (const float*)d_in[0];
    const long long* ei    = (const long long*)d_in[1];
    const long long* srcI  = ei;
    const long long* dstI  = ei + N_EDGES;
    const long long* batch = (const long long*)d_in[2];
    const float* W1r = (const float*)d_in[3];
    const float* b1  = (const float*)d_in[4];
    const float* W1t = (const float*)d_in[5];
    const float* W2r = (const float*)d_in[6];
    const float* b2  = (const float*)d_in[7];
    const float* W2t = (const float*)d_in[8];
    const float* W3r = (const float*)d_in[9];
    const float* b3  = (const float*)d_in[10];
    const float* W3t = (const float*)d_in[11];
    const float* Wl  = (const float*)d_in[12];
    const float* bl  = (const float*)d_in[13];
    float* out = (float*)d_out;

    char* ws = (char*)d_ws;
    size_t off = 0;
    float* agg = (float*)(ws + off); off += alignup((size_t)N_NODES * 32 * 4);
    float* h1  = (float*)(ws + off); off += alignup((size_t)N_NODES * 32 * 4);
    float* h2  = (float*)(ws + off); off += alignup((size_t)N_NODES * 16 * 4);
    float* h3  = (float*)(ws + off); off += alignup((size_t)N_NODES * 8 * 4);
    float* pooled = (float*)(ws + off);

    const int B = 256;
    #define GB(count) ((int)(((long long)(count) + B - 1) / B))

    // ---- layer 1: 3 -> 32 ----
    zero_f32<<<GB(N_NODES * 3), B, 0, stream>>>(agg, N_NODES * 3);
    scatter_add3<<<GB(N_EDGES), B, 0, stream>>>(x, srcI, dstI, agg, N_EDGES);
    {
        int waves = (N_NODES / 16) * 2;  // OUT=32 -> 2 col tiles
        transform_relu_wmma<3, 32><<<GB(waves * 32), B, 0, stream>>>(agg, x, W1r, b1, W1t, h1, N_NODES);
    }

    // ---- layer 2: 32 -> 16 ----
    zero_f32<<<GB(N_NODES * 32), B, 0, stream>>>(agg, N_NODES * 32);
    scatter_add_vec<32><<<GB((long long)N_EDGES * 8), B, 0, stream>>>(h1, srcI, dstI, agg, N_EDGES);
    {
        int waves = (N_NODES / 16) * 1;
        transform_relu_wmma<32, 16><<<GB(waves * 32), B, 0, stream>>>(agg, h1, W2r, b2, W2t, h2, N_NODES);
    }

    // ---- layer 3: 16 -> 8 ----
    zero_f32<<<GB(N_NODES * 16), B, 0, stream>>>(agg, N_NODES * 16);
    scatter_add_vec<16><<<GB((long long)N_EDGES * 4), B, 0, stream>>>(h2, srcI, dstI, agg, N_EDGES);
    {
        int waves = (N_NODES / 16) * 1;  // OUT=8 padded to one 16-wide tile
        transform_relu_wmma<16, 8><<<GB(waves * 32), B, 0, stream>>>(agg, h2, W3r, b3, W3t, h3, N_NODES);
    }

    // ---- pool + head ----
    zero_f32<<<GB(N_GRAPHS * 8), B, 0, stream>>>(pooled, N_GRAPHS * 8);
    pool_max8<<<GB(N_NODES * 8), B, 0, stream>>>(h3, batch, (unsigned int*)pooled, N_NODES);
    final_linear<<<GB(N_GRAPHS), B, 0, stream>>>(pooled, Wl, bl, out, N_GRAPHS);
    #undef GB
}